// ContrastiveModule_11785390260683
// MI455X (gfx1250) — compile-verified
//
#include <hip/hip_runtime.h>
#include <math.h>

typedef __attribute__((ext_vector_type(2))) float v2f;
typedef __attribute__((ext_vector_type(8))) float v8f;

#define HW   4096
#define CIN  256
#define CCH  32
#define NB   2
#define SLOT (NB * CCH * HW)   /* 262144 floats: one [b,c,h,w] tensor */
#define BCH  (CCH * HW)        /* 131072 floats: one batch of one tensor */

// ---------------------------------------------------------------------------
// 1x1 conv: Y[o,p] = sum_c W[o,c] * x[c,p]  for 9 (W,x) pairs x 2 batches.
// pp = wi*3 + xi;  pp<6 -> workspace (Q1..Q3,K1..K3), pp>=6 -> d_out V slots.
// ---------------------------------------------------------------------------
__global__ void conv_kernel(const float* __restrict__ x1, const float* __restrict__ x2,
                            const float* __restrict__ x3,
                            const float* __restrict__ W1, const float* __restrict__ W2,
                            const float* __restrict__ W3,
                            float* __restrict__ yqk, float* __restrict__ out) {
    const int pp = blockIdx.y, batch = blockIdx.z;
    const int wi = pp / 3, xi = pp % 3;
    const float* x = (xi == 0 ? x1 : (xi == 1 ? x2 : x3)) + (size_t)batch * CIN * HW;
    const float* W = (wi == 0 ? W1 : (wi == 1 ? W2 : W3));
    __shared__ float sW[CCH * CIN];
    for (int t = threadIdx.x; t < CCH * CIN; t += blockDim.x) sW[t] = W[t];
    __syncthreads();
    const int p = blockIdx.x * blockDim.x + threadIdx.x;
    float acc[CCH];
#pragma unroll
    for (int o = 0; o < CCH; ++o) acc[o] = 0.f;
    for (int ci = 0; ci < CIN; ++ci) {
        const float xv = x[(size_t)ci * HW + p];
#pragma unroll
        for (int o = 0; o < CCH; ++o) acc[o] = fmaf(sW[o * CIN + ci], xv, acc[o]);
    }
    float* Y = (pp < 6) ? (yqk + (size_t)pp * SLOT + (size_t)batch * BCH)
                        : (out + (size_t)(5 + pp - 6) * SLOT + (size_t)batch * BCH);
#pragma unroll
    for (int o = 0; o < CCH; ++o) Y[o * HW + p] = acc[o];
}

// ---------------------------------------------------------------------------
// Instance-norm statistics per (pair, batch, channel): mean and rsqrt(var+eps)
// ---------------------------------------------------------------------------
__global__ void stats_kernel(const float* __restrict__ yqk, const float* __restrict__ out,
                             float* __restrict__ mean, float* __restrict__ rstd) {
    const int o = blockIdx.x, pp = blockIdx.y, batch = blockIdx.z;
    const float* Y = ((pp < 6) ? (yqk + (size_t)pp * SLOT)
                               : (out + (size_t)(5 + pp - 6) * SLOT))
                     + (size_t)batch * BCH + (size_t)o * HW;
    float s = 0.f, q = 0.f;
    for (int p = threadIdx.x; p < HW; p += blockDim.x) {
        const float v = Y[p];
        s += v; q += v * v;
    }
    __shared__ float rs[256], rq[256];
    rs[threadIdx.x] = s; rq[threadIdx.x] = q;
    __syncthreads();
    for (int st = 128; st > 0; st >>= 1) {
        if ((int)threadIdx.x < st) {
            rs[threadIdx.x] += rs[threadIdx.x + st];
            rq[threadIdx.x] += rq[threadIdx.x + st];
        }
        __syncthreads();
    }
    if (threadIdx.x == 0) {
        const float mu  = rs[0] * (1.f / HW);
        const float var = rq[0] * (1.f / HW) - mu * mu;
        const int idx = (pp * NB + batch) * CCH + o;
        mean[idx] = mu;
        rstd[idx] = rsqrtf(var + 1e-5f);
    }
}

// ---------------------------------------------------------------------------
// Normalize + affine + ReLU in place (workspace for Q/K, d_out for V).
// ---------------------------------------------------------------------------
__global__ void norm_kernel(float* __restrict__ yqk, float* __restrict__ out,
                            const float* __restrict__ mean, const float* __restrict__ rstd,
                            const float* __restrict__ g1, const float* __restrict__ b1,
                            const float* __restrict__ g2, const float* __restrict__ b2,
                            const float* __restrict__ g3, const float* __restrict__ b3) {
    const int pp = blockIdx.y, batch = blockIdx.z;
    const int wi = pp / 3;
    const float* g  = (wi == 0 ? g1 : (wi == 1 ? g2 : g3));
    const float* be = (wi == 0 ? b1 : (wi == 1 ? b2 : b3));
    float* Y = ((pp < 6) ? (yqk + (size_t)pp * SLOT)
                         : (out + (size_t)(5 + pp - 6) * SLOT))
               + (size_t)batch * BCH;
    const int p = blockIdx.x * blockDim.x + threadIdx.x;
    const int sb = (pp * NB + batch) * CCH;
#pragma unroll 4
    for (int o = 0; o < CCH; ++o) {
        float v = (Y[o * HW + p] - mean[sb + o]) * rstd[sb + o] * g[o] + be[o];
        Y[o * HW + p] = v > 0.f ? v : 0.f;
    }
}

// ---------------------------------------------------------------------------
// Pass 1: per-row softmax stats of S = Q^T K (S[i,j] = sum_c Q[c,i]K[c,j]).
// One wave per 16-row tile.  WMMA f32 16x16x4, K accumulated over 8 steps.
// Each lane keeps exact (m,l) stats over its own disjoint column partition
// (C/D layout: lane holds col N = lane%16, rows r+8*half), merged at the end
// with a 4-step xor-shuffle within each 16-lane half.  Single GEMM sweep.
// ---------------------------------------------------------------------------
__global__ void pass1_kernel(const float* __restrict__ Q, const float* __restrict__ K,
                             float* __restrict__ rowM, float* __restrict__ rowZ) {
    const int lane = threadIdx.x;
    const int half = lane >> 4;
    const int l16  = lane & 15;
    const int i0   = blockIdx.x * 16;
    const int batch = blockIdx.y;
    const float* Qb = Q + (size_t)batch * BCH;
    const float* Kb = K + (size_t)batch * BCH;

    v2f a[8];                                  // A frags: 16 rows x K=32
#pragma unroll
    for (int kk = 0; kk < 8; ++kk) {
        const int c0 = kk * 4 + half * 2;      // ISA fp32 A layout: VGPR0=K0/K2, VGPR1=K1/K3
        a[kk][0] = Qb[(c0 + 0) * HW + i0 + l16];
        a[kk][1] = Qb[(c0 + 1) * HW + i0 + l16];
    }
    float m[8], l[8];
#pragma unroll
    for (int r = 0; r < 8; ++r) { m[r] = -INFINITY; l[r] = 0.f; }

    for (int jt = 0; jt < HW / 16; ++jt) {
        const int j0 = jt * 16;
        v8f c = {0.f, 0.f, 0.f, 0.f, 0.f, 0.f, 0.f, 0.f};
#pragma unroll
        for (int kk = 0; kk < 8; ++kk) {
            const int c0 = kk * 4 + half * 2;
            v2f bf;
            bf[0] = Kb[(c0 + 0) * HW + j0 + l16];
            bf[1] = Kb[(c0 + 1) * HW + j0 + l16];
            c = __builtin_amdgcn_wmma_f32_16x16x4_f32(false, a[kk], false, bf,
                                                      (short)0, c, false, false);
        }
#pragma unroll
        for (int r = 0; r < 8; ++r) {
            const float s  = c[r];
            const float mn = fmaxf(m[r], s);
            l[r] = l[r] * __expf(m[r] - mn) + __expf(s - mn);
            m[r] = mn;
        }
    }
#pragma unroll
    for (int r = 0; r < 8; ++r) {
        float mm = m[r], ll = l[r];
#pragma unroll
        for (int off = 1; off < 16; off <<= 1) {   // merge (m,l) pairs within half
            const float mo = __shfl_xor(mm, off, 16);
            const float lo = __shfl_xor(ll, off, 16);
            const float mn = fmaxf(mm, mo);
            ll = ll * __expf(mm - mn) + lo * __expf(mo - mn);
            mm = mn;
        }
        if (l16 == 0) {
            const int row = i0 + r + half * 8;
            rowM[batch * HW + row] = mm;
            rowZ[batch * HW + row] = ll;
        }
    }
}

// ---------------------------------------------------------------------------
// alpha_i = w_i / Z_i  (w == nullptr means ones -> colsum)
// ---------------------------------------------------------------------------
__global__ void prep_alpha(const float* w1, const float* w2, const float* __restrict__ Z,
                           float* __restrict__ a1, float* __restrict__ a2) {
    const int i = blockIdx.x * blockDim.x + threadIdx.x;
    const float rz = 1.f / Z[i];
    a1[i] = (w1 ? w1[i] : 1.f) * rz;
    a2[i] = (w2 ? w2[i] : 1.f) * rz;
}

// ---------------------------------------------------------------------------
// Pass 2: out[j] = sum_i alpha_i * exp(S[i,j] - m_i), up to two alpha vectors
// at once.  One wave per 16-column tile; B frags (K cols) held in registers,
// A frags (Q rows) streamed; stats broadcast via intra-half shuffles.
// ---------------------------------------------------------------------------
__global__ void pass2_kernel(const float* __restrict__ Q, const float* __restrict__ K,
                             const float* __restrict__ rowM,
                             const float* __restrict__ a1, const float* __restrict__ a2,
                             float* __restrict__ o1, float* __restrict__ o2, int has2) {
    const int lane = threadIdx.x;
    const int half = lane >> 4;
    const int l16  = lane & 15;
    const int j0   = blockIdx.x * 16;
    const int batch = blockIdx.y;
    const float* Qb  = Q + (size_t)batch * BCH;
    const float* Kb  = K + (size_t)batch * BCH;
    const float* Mb  = rowM + batch * HW;
    const float* A1b = a1 + batch * HW;
    const float* A2b = a2 + batch * HW;

    v2f bf[8];                                 // B frags for this column tile
#pragma unroll
    for (int kk = 0; kk < 8; ++kk) {
        const int c0 = kk * 4 + half * 2;
        bf[kk][0] = Kb[(c0 + 0) * HW + j0 + l16];
        bf[kk][1] = Kb[(c0 + 1) * HW + j0 + l16];
    }
    float acc1 = 0.f, acc2 = 0.f;
    for (int it = 0; it < HW / 16; ++it) {
        const int i0 = it * 16;
        const float mv  = Mb[i0 + l16];        // lane caches stats for row i0+l16
        const float a1v = A1b[i0 + l16];
        const float a2v = has2 ? A2b[i0 + l16] : 0.f;
        v8f c = {0.f, 0.f, 0.f, 0.f, 0.f, 0.f, 0.f, 0.f};
#pragma unroll
        for (int kk = 0; kk < 8; ++kk) {
            const int c0 = kk * 4 + half * 2;
            v2f af;
            af[0] = Qb[(c0 + 0) * HW + i0 + l16];
            af[1] = Qb[(c0 + 1) * HW + i0 + l16];
            c = __builtin_amdgcn_wmma_f32_16x16x4_f32(false, af, false, bf[kk],
                                                      (short)0, c, false, false);
        }
#pragma unroll
        for (int r = 0; r < 8; ++r) {
            const int t = r + half * 8;        // row-in-tile this lane's c[r] covers
            const float e = __expf(c[r] - __shfl(mv, t, 16));
            acc1 += __shfl(a1v, t, 16) * e;
            if (has2) acc2 += __shfl(a2v, t, 16) * e;
        }
    }
    acc1 += __shfl_xor(acc1, 16, 32);          // halves covered disjoint row sets
    acc2 += __shfl_xor(acc2, 16, 32);
    if (lane < 16) {
        o1[batch * HW + j0 + l16] = acc1;
        if (has2) o2[batch * HW + j0 + l16] = acc2;
    }
}

// ---------------------------------------------------------------------------
// Broadcast five [b,hw] vectors to [b,c,h,w] output slots 0..4.
// ---------------------------------------------------------------------------
__global__ void bcast_kernel(const float* __restrict__ p1a, const float* __restrict__ p1b,
                             const float* __restrict__ p2v, const float* __restrict__ p3a,
                             const float* __restrict__ p3b, float* __restrict__ out) {
    const int slot = blockIdx.y, batch = blockIdx.z;
    const float* v = (slot == 0) ? p1a : (slot == 1) ? p1b : (slot == 2) ? p2v
                   : (slot == 3) ? p3a : p3b;
    const int p = blockIdx.x * blockDim.x + threadIdx.x;
    const float val = v[batch * HW + p];
    float* o = out + (size_t)slot * SLOT + (size_t)batch * BCH;
#pragma unroll 4
    for (int ch = 0; ch < CCH; ++ch) o[ch * HW + p] = val;
}

// ---------------------------------------------------------------------------
extern "C" void kernel_launch(void* const* d_in, const int* in_sizes, int n_in,
                              void* d_out, int out_size, void* d_ws, size_t ws_size,
                              hipStream_t stream) {
    (void)in_sizes; (void)n_in; (void)out_size; (void)ws_size;
    const float* x1 = (const float*)d_in[0];
    const float* x2 = (const float*)d_in[1];
    const float* x3 = (const float*)d_in[2];
    const float* W1 = (const float*)d_in[3];
    const float* g1 = (const float*)d_in[4];
    const float* b1 = (const float*)d_in[5];
    const float* W2 = (const float*)d_in[6];
    const float* g2 = (const float*)d_in[7];
    const float* b2 = (const float*)d_in[8];
    const float* W3 = (const float*)d_in[9];
    const float* g3 = (const float*)d_in[10];
    const float* b3 = (const float*)d_in[11];
    float* out = (float*)d_out;
    float* ws  = (float*)d_ws;

    // workspace layout (floats); total ~1.68M floats (~6.7 MB)
    float* YQK  = ws;                          // 6 * SLOT = 1,572,864
    float* rowM = YQK + 6 * SLOT;              // NB*HW
    float* rowZ = rowM + NB * HW;
    float* A1   = rowZ + NB * HW;
    float* A2   = A1 + NB * HW;
    float* MEAN = A2 + NB * HW;                // 9*NB*CCH = 576
    float* RSTD = MEAN + 9 * NB * CCH;
    float* c12  = RSTD + 9 * NB * CCH;
    float* c23  = c12 + NB * HW;
    float* t2   = c23 + NB * HW;               // == p3a
    float* t3   = t2 + NB * HW;
    float* p1a  = t3 + NB * HW;
    float* p1b  = p1a + NB * HW;
    float* p2v  = p1b + NB * HW;
    float* p3b  = p2v + NB * HW;

    // branches
    conv_kernel <<<dim3(HW / 256, 9, NB), 256, 0, stream>>>(x1, x2, x3, W1, W2, W3, YQK, out);
    stats_kernel<<<dim3(CCH, 9, NB),      256, 0, stream>>>(YQK, out, MEAN, RSTD);
    norm_kernel <<<dim3(HW / 256, 9, NB), 256, 0, stream>>>(YQK, out, MEAN, RSTD,
                                                            g1, b1, g2, b2, g3, b3);
    const float* Q1 = YQK + 0 * SLOT;
    const float* Q2 = YQK + 1 * SLOT;
    const float* Q3 = YQK + 2 * SLOT;
    const float* K1 = YQK + 3 * SLOT;
    const float* K2 = YQK + 4 * SLOT;
    const float* K3 = YQK + 5 * SLOT;

    const dim3 tg(HW / 16, NB);
    auto runA = [&](const float* Q, const float* K, const float* w1p, const float* w2p,
                    float* o1, float* o2) {
        pass1_kernel<<<tg, 32, 0, stream>>>(Q, K, rowM, rowZ);
        prep_alpha<<<(NB * HW) / 256, 256, 0, stream>>>(w1p, w2p, rowZ, A1, A2);
        pass2_kernel<<<tg, 32, 0, stream>>>(Q, K, rowM, A1, A2, o1, o2, o2 != nullptr);
    };
    runA(Q1, K2, nullptr, nullptr, c12, nullptr);   // A12: colsum -> c12
    runA(Q1, K3, nullptr, nullptr, p3b, nullptr);   // A13: colsum -> p3b
    runA(Q2, K3, nullptr, c12,     c23, t2);        // A23: colsum -> c23 ; c12@A23 -> t2 (=p3a)
    runA(Q3, K2, c23,     t2,      p2v, t3);        // A32: c23@A32 -> p2 ; t2@A32 -> t3
    runA(Q2, K1, c12,     t3,      p1b, p1a);       // A21: c12@A21 -> p1b ; t3@A21 -> p1a

    bcast_kernel<<<dim3(HW / 256, 5, NB), 256, 0, stream>>>(p1a, p1b, p2v, t2, p3b, out);
}